// HollywoodTritonFused_53584011985644
// MI455X (gfx1250) — compile-verified
//
#include <hip/hip_runtime.h>
#include <stdint.h>

// ---------------------------------------------------------------------------
// Batched 4096-point complex FFT (split re/im, fp32), one row per workgroup.
//
// Roofline: 512 MiB of HBM traffic @ 23.3 TB/s ~= 22 us; ~2 GFLOP of butterfly
// work. Memory-bound => single pass over global data, whole row resident in
// LDS/registers. 4096 = 16*16*16, three radix-16 passes, two LDS exchanges.
// Pass A (DFT-16 over stride-256 columns) is done as F16(16x16) x X(16x256)
// complex matmul on the matrix pipe via V_WMMA_F32_16X16X4_F32 chains.
// Global->LDS staging uses the Tensor Data Mover (TENSOR_LOAD_TO_LDS).
// ---------------------------------------------------------------------------

#define FFT_N       4096
#define ROW_THREADS 256
#define EX_STRIDE   272   // 256 + 16 pad: row offset == 16 (mod 64 banks)

typedef float v2f __attribute__((ext_vector_type(2)));
typedef float v8f __attribute__((ext_vector_type(8)));
typedef unsigned int v4u __attribute__((ext_vector_type(4)));
typedef int v4i __attribute__((ext_vector_type(4)));
typedef int v8i __attribute__((ext_vector_type(8)));

#if defined(__has_builtin)
#  if __has_builtin(__builtin_amdgcn_tensor_load_to_lds)
#    define HAVE_TDM 1
#  endif
#  if __has_builtin(__builtin_amdgcn_wmma_f32_16x16x4_f32)
#    define HAVE_WMMA4 1
#  endif
#  if __has_builtin(__builtin_amdgcn_s_wait_tensorcnt)
#    define HAVE_WAIT_TENSOR 1
#  endif
#endif
#ifndef HAVE_TDM
#  define HAVE_TDM 0
#endif
#ifndef HAVE_WMMA4
#  define HAVE_WMMA4 0
#endif
#ifndef HAVE_WAIT_TENSOR
#  define HAVE_WAIT_TENSOR 0
#endif

#if __has_include(<hip/amd_detail/amd_gfx1250_TDM.h>)
#  define TDM_SIX_ARGS 1   // amdgpu-toolchain (clang-23): 6-arg builtin
#else
#  define TDM_SIX_ARGS 0   // ROCm 7.2 (clang-22): 5-arg builtin
#endif

// twiddle constants
#define TW4096f (-1.53398078e-3f)   // -2*pi/4096
#define TW256f  (-2.45436926e-2f)   // -2*pi/256
#define TW16f   (-3.92699082e-1f)   // -2*pi/16

// W16^j = exp(-i*pi*j/8), j = 0..7
constexpr float kW16C[8] = { 1.0f,  0.92387953f,  0.70710678f,  0.38268343f,
                             0.0f, -0.38268343f, -0.70710678f, -0.92387953f };
constexpr float kW16S[8] = { 0.0f, -0.38268343f, -0.70710678f, -0.92387953f,
                            -1.0f, -0.92387953f, -0.70710678f, -0.38268343f };
// 4-bit bit reversal
constexpr int kBR4[16] = {0,8,4,12,2,10,6,14,1,9,5,13,3,11,7,15};

// In-register 16-point DIF FFT: natural-order input, output[i] = X[brev4(i)].
__device__ __forceinline__ void dft16_dif(float re[16], float im[16]) {
#pragma unroll
  for (int h = 8; h >= 1; h >>= 1) {
#pragma unroll
    for (int g = 0; g < 16; g += 2 * h) {
#pragma unroll
      for (int j = 0; j < h; ++j) {
        const int a = g + j, b = a + h;
        const float ur = re[a], ui = im[a];
        const float vr = re[b], vi = im[b];
        re[a] = ur + vr;
        im[a] = ui + vi;
        const float dr = ur - vr, di = ui - vi;
        const int ti = j * (8 / h);         // W_{2h}^j = W16^{j*8/h}
        const float wr = kW16C[ti], wi = kW16S[ti];
        re[b] = dr * wr - di * wi;
        im[b] = dr * wi + di * wr;
      }
    }
  }
}

#if HAVE_TDM
// TDM: DMA one contiguous 4096-float row from global into LDS.
__device__ __forceinline__ void tdm_load_row(const float* gptr, float* lptr) {
  const unsigned lds = (unsigned)(uintptr_t)lptr;          // addr[31:0] = LDS offset
  const unsigned long long ga = (unsigned long long)(uintptr_t)gptr;

  v4u g0;                                                  // D# group 0 (128b)
  g0.x = 1u;                                               // count=1, user mode
  g0.y = lds;                                              // lds_addr
  g0.z = (unsigned)(ga & 0xffffffffull);                   // global_addr[31:0]
  g0.w = (unsigned)((ga >> 32) & 0x01ffffffull)            // global_addr[56:32]
       | 0x80000000u;                                      // type=2 ("image")

  v8i g1;                                                  // D# group 1 (256b)
  g1[0] = 0x00020000;                 // workgroup_mask=0, data_size=2 (4B)
  g1[1] = (int)(0x1000u << 16);       // tensor_dim0 = 4096 (bits 63:48)
  g1[2] = (int)(1u << 16);            // tensor_dim0 hi=0, tensor_dim1 = 1
  g1[3] = (int)(0x1000u << 16);       // tensor_dim1 hi=0, tile_dim0 = 4096
  g1[4] = 1;                          // tile_dim1 = 1, tile_dim2 = 0
  g1[5] = 4096;                       // tensor_dim0_stride lo32
  g1[6] = (int)(0x1000u << 16);       // stride0 hi=0, tensor_dim1_stride lo16
  g1[7] = 0;

  v4i z4 = {0, 0, 0, 0};
#if TDM_SIX_ARGS
  v8i z8 = {0, 0, 0, 0, 0, 0, 0, 0};
  __builtin_amdgcn_tensor_load_to_lds(g0, g1, z4, z4, z8, 0);
#else
  __builtin_amdgcn_tensor_load_to_lds(g0, g1, z4, z4, 0);
#endif
}
#endif

__global__ __launch_bounds__(ROW_THREADS) void fft4096_rows_kernel(
    const float* __restrict__ xre, const float* __restrict__ xim,
    float* __restrict__ outre, float* __restrict__ outim) {
  __shared__ float sRe[FFT_N];               // TDM staging (natural order)
  __shared__ float sIm[FFT_N];
  __shared__ float eRe[16 * EX_STRIDE];      // exchange matrix, 16 rows
  __shared__ float eIm[16 * EX_STRIDE];

  const int row = blockIdx.x;
  const int tid = threadIdx.x;
  const size_t base = (size_t)row * FFT_N;

  // ---- Stage the row into LDS --------------------------------------------
#if HAVE_TDM
  if (tid < 32) {                            // wave 0 issues the DMAs
    tdm_load_row(xre + base, sRe);
    tdm_load_row(xim + base, sIm);
#if HAVE_WAIT_TENSOR
    __builtin_amdgcn_s_wait_tensorcnt(0);
#else
    asm volatile("s_wait_tensorcnt 0" ::: "memory");
#endif
  }
#else
#pragma unroll
  for (int j = 0; j < 16; ++j) {
    sRe[tid + 256 * j] = xre[base + tid + 256 * j];
    sIm[tid + 256 * j] = xim[base + tid + 256 * j];
  }
#endif
  __syncthreads();

  const int lane = tid & 31;
  const int wv = tid >> 5;
  const int ln = lane & 15;
  const int lh = lane >> 4;                  // half-wave select

  // ---- Pass A: G[k1][n2] = sum_n1 x[n1*256+n2] * W16^{n1*k1},
  //      then twiddle by W4096^{n2*k1}; store to exchange rows (stride 272).
#if HAVE_WMMA4
  {
    // Per-lane DFT-16 matrix chunks in WMMA A layout (16x4 per instruction):
    // lanes 0-15 hold M=k1, VGPR pair covers K=(0,1) / lanes 16-31 K=(2,3).
    v2f Ar[4], Ai[4], Ani[4];
#pragma unroll
    for (int kk = 0; kk < 4; ++kk) {
      const int k0 = 4 * kk + 2 * lh;
      float s0, c0, s1, c1;
      __sincosf(TW16f * (float)((ln * k0) & 15), &s0, &c0);
      __sincosf(TW16f * (float)((ln * (k0 + 1)) & 15), &s1, &c1);
      v2f a;  a.x = c0;  a.y = c1;  Ar[kk] = a;
      v2f b;  b.x = s0;  b.y = s1;  Ai[kk] = b;
      v2f n;  n.x = -s0; n.y = -s1; Ani[kk] = n;   // f32 WMMA has no A-negate
    }
#pragma unroll
    for (int rep = 0; rep < 2; ++rep) {      // 8 waves x 2 tiles = 16 tiles
      const int c0 = wv * 32 + rep * 16;
      const int nc = c0 + ln;                // this lane's column n2
      v2f Br[4], Bi[4];
#pragma unroll
      for (int kk = 0; kk < 4; ++kk) {
        const int rA = 4 * kk + 2 * lh;
        v2f br; br.x = sRe[rA * 256 + nc]; br.y = sRe[(rA + 1) * 256 + nc];
        v2f bi; bi.x = sIm[rA * 256 + nc]; bi.y = sIm[(rA + 1) * 256 + nc];
        Br[kk] = br; Bi[kk] = bi;
      }
      v8f dr = {0, 0, 0, 0, 0, 0, 0, 0};
      v8f di = {0, 0, 0, 0, 0, 0, 0, 0};
#pragma unroll
      for (int kk = 0; kk < 4; ++kk)         // Re += Fr*Xr
        dr = __builtin_amdgcn_wmma_f32_16x16x4_f32(false, Ar[kk], false, Br[kk],
                                                   (short)0, dr, false, false);
#pragma unroll
      for (int kk = 0; kk < 4; ++kk)         // Re += (-Fi)*Xi
        dr = __builtin_amdgcn_wmma_f32_16x16x4_f32(false, Ani[kk], false, Bi[kk],
                                                   (short)0, dr, false, false);
#pragma unroll
      for (int kk = 0; kk < 4; ++kk)         // Im += Fi*Xr
        di = __builtin_amdgcn_wmma_f32_16x16x4_f32(false, Ai[kk], false, Br[kk],
                                                   (short)0, di, false, false);
#pragma unroll
      for (int kk = 0; kk < 4; ++kk)         // Im += Fr*Xi
        di = __builtin_amdgcn_wmma_f32_16x16x4_f32(false, Ar[kk], false, Bi[kk],
                                                   (short)0, di, false, false);
      // D layout: lane n holds column; VGPR v is row v (+8 for upper lanes).
#pragma unroll
      for (int v = 0; v < 8; ++v) {
        const int k1 = v + 8 * lh;
        float s, c;
        __sincosf(TW4096f * (float)(nc * k1), &s, &c);
        const float gr = dr[v], gi = di[v];
        eRe[k1 * EX_STRIDE + nc] = gr * c - gi * s;
        eIm[k1 * EX_STRIDE + nc] = gr * s + gi * c;
      }
    }
  }
#else
  {
    float rr[16], ri[16];
#pragma unroll
    for (int i = 0; i < 16; ++i) {
      rr[i] = sRe[i * 256 + tid];
      ri[i] = sIm[i * 256 + tid];
    }
    dft16_dif(rr, ri);
#pragma unroll
    for (int i = 0; i < 16; ++i) {
      const int k1 = kBR4[i];
      float s, c;
      __sincosf(TW4096f * (float)(tid * k1), &s, &c);
      eRe[k1 * EX_STRIDE + tid] = rr[i] * c - ri[i] * s;
      eIm[k1 * EX_STRIDE + tid] = rr[i] * s + ri[i] * c;
    }
  }
#endif
  __syncthreads();

  // ---- Pass B: per (k1, n2'): DFT-16 over n1', twiddle W256^{n2'*k1''}.
  // Row k1 is touched only by its own 16 lanes (one wave) -> no barrier
  // needed between the in-row read and write (LDS is in-order per wave).
  {
    const int k1 = tid >> 4;
    const int np = tid & 15;
    float rr[16], ri[16];
#pragma unroll
    for (int j = 0; j < 16; ++j) {
      rr[j] = eRe[k1 * EX_STRIDE + j * 16 + np];
      ri[j] = eIm[k1 * EX_STRIDE + j * 16 + np];
    }
    dft16_dif(rr, ri);
#pragma unroll
    for (int i = 0; i < 16; ++i) {
      const int kpp = kBR4[i];
      float s, c;
      __sincosf(TW256f * (float)(np * kpp), &s, &c);
      const float tr = rr[i] * c - ri[i] * s;
      const float ti = rr[i] * s + ri[i] * c;
      eRe[k1 * EX_STRIDE + np * 17 + kpp] = tr;   // stride-17 conflict-free pack
      eIm[k1 * EX_STRIDE + np * 17 + kpp] = ti;
    }
  }
  __syncthreads();

  // ---- Pass C: per (k1, k1''): DFT-16 over n2' (no twiddle), then place
  // into swizzled output layout for coalesced stores.
  {
    const int k1 = tid >> 4;
    const int kpp = tid & 15;
    float rr[16], ri[16];
#pragma unroll
    for (int j = 0; j < 16; ++j) {
      rr[j] = eRe[k1 * EX_STRIDE + j * 17 + kpp];
      ri[j] = eIm[k1 * EX_STRIDE + j * 17 + kpp];
    }
    __syncthreads();                       // writes go to all rows -> barrier
    dft16_dif(rr, ri);
#pragma unroll
    for (int i = 0; i < 16; ++i) {
      const int k2 = kBR4[i];
      // output position p = k1 + 16*k1'' + 256*k2 stored at swizzled addr
      eRe[k2 * EX_STRIDE + k1 * 17 + kpp] = rr[i];
      eIm[k2 * EX_STRIDE + k1 * 17 + kpp] = ri[i];
    }
  }
  __syncthreads();

  // ---- Coalesced output stores (128B per wave instruction) ---------------
#pragma unroll
  for (int j = 0; j < 16; ++j) {
    const int p = tid + 256 * j;
    const int a = (p >> 8) * EX_STRIDE + (p & 15) * 17 + ((p >> 4) & 15);
    outre[base + p] = eRe[a];
    outim[base + p] = eIm[a];
  }
}

extern "C" void kernel_launch(void* const* d_in, const int* in_sizes, int n_in,
                              void* d_out, int out_size, void* d_ws, size_t ws_size,
                              hipStream_t stream) {
  (void)n_in; (void)d_ws; (void)ws_size;
  const float* xre = (const float*)d_in[0];
  const float* xim = (const float*)d_in[1];
  const int rows = in_sizes[0] / FFT_N;          // BATCH
  float* outre = (float*)d_out;                  // y_re, then y_im
  float* outim = (float*)d_out + (size_t)rows * FFT_N;
  (void)out_size;
  fft4096_rows_kernel<<<dim3(rows), dim3(ROW_THREADS), 0, stream>>>(
      xre, xim, outre, outim);
}